// gLASE_55009941127296
// MI455X (gfx1250) — compile-verified
//
#include <hip/hip_runtime.h>

typedef float v2f __attribute__((ext_vector_type(2)));
typedef float v8f __attribute__((ext_vector_type(8)));

#define NNODES 50000
#define NEDGES 800000
#define DCH    64

// Guaranteed hardware FP32 atomic add (no CAS loop), device scope.
__device__ __forceinline__ void hw_atomic_add_f32(float* p, float v) {
    asm volatile("global_atomic_add_f32 %0, %1, off" :: "v"(p), "v"(v) : "memory");
}

// ---------------------------------------------------------------------------
// Zero the two accumulator buffers.
__global__ void zero2_kernel(float* __restrict__ a, float* __restrict__ b, int n) {
    int i = blockIdx.x * blockDim.x + threadIdx.x;
    if (i < n) { a[i] = 0.0f; b[i] = 0.0f; }
}

// ---------------------------------------------------------------------------
// x_agg[dst] += x[src]  : one wave per edge, lane l handles channels 2l,2l+1.
__global__ void edge_agg_kernel(const float* __restrict__ x,
                                const int* __restrict__ ei,   // [2,E]
                                float* __restrict__ xagg) {
    int e = blockIdx.x * 8 + (threadIdx.x >> 5);
    if (e >= NEDGES) return;
    int lane = threadIdx.x & 31;
    int s = ei[e];
    int d = ei[NEDGES + e];
    v2f v = *(const v2f*)(x + (size_t)s * DCH + lane * 2);
    float* p = xagg + (size_t)d * DCH + lane * 2;
    hw_atomic_add_f32(p,     v.x);
    hw_atomic_add_f32(p + 1, v.y);
}

// ---------------------------------------------------------------------------
// Dual GEMM with V_WMMA_F32_16X16X4_F32 (exact fp32 path):
//   Y1 = Xa @ W1^T   (waves 0..3, each owns N-tile n0 = wave*16)
//   Y2 = X  @ W2^T   (waves 4..7)
// One block = 16 rows (M-tile). A layout: lane holds row (lane&15),
// K pair offset (lane>>4)*2. B[k][o] = W[o][k] -> same half-lane K split.
__global__ void gemm2_wmma_kernel(const float* __restrict__ Xa,
                                  const float* __restrict__ X,
                                  const float* __restrict__ W1,
                                  const float* __restrict__ W2,
                                  float* __restrict__ Y1,
                                  float* __restrict__ Y2) {
    int m0    = blockIdx.x * 16;
    int wave  = threadIdx.x >> 5;     // 0..7
    int lane  = threadIdx.x & 31;
    int which = wave >> 2;            // 0 -> (Xa,W1,Y1), 1 -> (X,W2,Y2)
    int n0    = (wave & 3) * 16;

    const float* A = which ? X  : Xa;
    const float* W = which ? W2 : W1;
    float*       Y = which ? Y2 : Y1;

    int half = lane & 15;             // row index within tile (A) / col (B)
    int kk   = (lane >> 4) * 2;       // K sub-offset for this lane half

    const float* arow = A + (size_t)(m0 + half) * DCH + kk;  // A[m][k]
    const float* brow = W + (size_t)(n0 + half) * DCH + kk;  // B[k][n] = W[n][k]

    v8f c = {};
#pragma unroll
    for (int k0 = 0; k0 < DCH; k0 += 4) {
        v2f a = *(const v2f*)(arow + k0);
        v2f b = *(const v2f*)(brow + k0);
        c = __builtin_amdgcn_wmma_f32_16x16x4_f32(
                /*neg_a=*/false, a, /*neg_b=*/false, b,
                /*c_mod=*/(short)0, c, /*reuse_a=*/false, /*reuse_b=*/false);
    }

    // C layout: VGPR i -> row m0 + i + (lane>=16 ? 8 : 0), col n0 + (lane&15)
    int rbase = m0 + ((lane >> 4) << 3);
    float* out = Y + n0 + half;
#pragma unroll
    for (int i = 0; i < 8; ++i)
        out[(size_t)(rbase + i) * DCH] = c[i];
}

// ---------------------------------------------------------------------------
// Fused per-edge attention + scatter:
//   attn = sum_d (x[s,d]*Q[d]*x[t,d]);  acc[t] += attn * x2[s]
// One wave per edge; wave32 xor-shuffle reduction.
__global__ void edge_attn_kernel(const float* __restrict__ x,
                                 const float* __restrict__ x2,
                                 const float* __restrict__ Q,
                                 const int* __restrict__ ei,   // [2,E]
                                 float* __restrict__ acc) {
    int e = blockIdx.x * 8 + (threadIdx.x >> 5);
    if (e >= NEDGES) return;
    int lane = threadIdx.x & 31;
    int s = ei[e];
    int d = ei[NEDGES + e];

    v2f xs = *(const v2f*)(x + (size_t)s * DCH + lane * 2);
    v2f xd = *(const v2f*)(x + (size_t)d * DCH + lane * 2);
    v2f q  = *(const v2f*)(Q + lane * 2);

    float p = xs.x * q.x * xd.x + xs.y * q.y * xd.y;
#pragma unroll
    for (int off = 16; off > 0; off >>= 1)
        p += __shfl_xor(p, off, 32);

    v2f v = *(const v2f*)(x2 + (size_t)s * DCH + lane * 2);
    float* pd = acc + (size_t)d * DCH + lane * 2;
    hw_atomic_add_f32(pd,     p * v.x);
    hw_atomic_add_f32(pd + 1, p * v.y);
}

// ---------------------------------------------------------------------------
// x_next = x + (y1 - acc) * Q
__global__ void combine_kernel(const float* __restrict__ x,
                               const float* __restrict__ y1,
                               const float* __restrict__ acc,
                               const float* __restrict__ Q,
                               float* __restrict__ out, int n) {
    int i = blockIdx.x * blockDim.x + threadIdx.x;
    if (i < n) {
        float q = Q[i & (DCH - 1)];
        out[i] = x[i] + (y1[i] - acc[i]) * q;
    }
}

// ---------------------------------------------------------------------------
extern "C" void kernel_launch(void* const* d_in, const int* in_sizes, int n_in,
                              void* d_out, int out_size, void* d_ws, size_t ws_size,
                              hipStream_t stream) {
    const float* x_in = (const float*)d_in[0];
    const int*   ei1  = (const int*)d_in[1];
    const int*   ei2  = (const int*)d_in[2];
    const float* W1   = (const float*)d_in[3];
    const float* W2   = (const float*)d_in[4];
    const float* Q    = (const float*)d_in[5];

    const size_t ND = (size_t)NNODES * DCH;
    float* ws   = (float*)d_ws;
    float* xbuf = ws;                 // ping-pong x (in-place safe across steps)
    float* xagg = ws + ND;
    float* y1   = ws + 2 * ND;
    float* x2   = ws + 3 * ND;
    float* acc  = ws + 4 * ND;

    const int elemBlocks = (int)((ND + 255) / 256);
    const int edgeBlocks = NEDGES / 8;

    const float* xcur = x_in;
    for (int t = 0; t < 3; ++t) {
        float* xnext = (t == 2) ? (float*)d_out : xbuf;
        const float* W1t = W1 + (size_t)t * DCH * DCH;
        const float* W2t = W2 + (size_t)t * DCH * DCH;
        const float* Qt  = Q  + (size_t)t * DCH;

        zero2_kernel<<<elemBlocks, 256, 0, stream>>>(xagg, acc, (int)ND);
        edge_agg_kernel<<<edgeBlocks, 256, 0, stream>>>(xcur, ei1, xagg);
        gemm2_wmma_kernel<<<NNODES / 16, 256, 0, stream>>>(xagg, xcur, W1t, W2t, y1, x2);
        edge_attn_kernel<<<edgeBlocks, 256, 0, stream>>>(xcur, x2, Qt, ei2, acc);
        combine_kernel<<<elemBlocks, 256, 0, stream>>>(xcur, y1, acc, Qt, xnext, (int)ND);
        xcur = xnext;
    }
}